// E8Quantizer_26010321944965
// MI455X (gfx1250) — compile-verified
//
#include <hip/hip_runtime.h>

typedef float v2f __attribute__((ext_vector_type(2)));
typedef float v8f __attribute__((ext_vector_type(8)));

#define NCODE  240
#define NTILE  15                 // 240/16 column tiles in pass 1
#define CBS    10                 // codebook row stride; col 8 = 1.0 (row-sum trick), col 9 = 0
#define WSP    20                 // W^T row stride (dwords): 16B aligned, bank-conflict-free
#define NCHUNK 60                 // 240/4 K-chunks in pass 2
#define SOFT_SCALE (3.3333333333333335f * 1.4426950408889634f)  // (1/T) * log2(e)

__global__ __launch_bounds__(64)
void e8_quant_wmma(const float* __restrict__ x,
                   const float* __restrict__ cb,
                   const float* __restrict__ cn,
                   float* __restrict__ out,
                   int nTiles, int nWaves)
{
    __shared__ float  s_cb[NCODE * CBS];      // 9.6 KB codebook (+ones column)
    __shared__ float  s_cn[NCODE];            // ||c||^2 * SOFT_SCALE
    __shared__ float  s_wt[2][NCODE * WSP];   // per-wave W^T [codeword][point]
    __shared__ float4 s_st[2][32];            // per-wave output staging

    const int tid = threadIdx.x;
    for (int i = tid; i < NCODE * CBS; i += 64) {
        const int row = i / CBS;
        const int col = i - row * CBS;
        float v = 0.0f;
        if (col < 8)       v = cb[row * 8 + col];
        else if (col == 8) v = 1.0f;          // ones column: row sums ride along in pass 2
        s_cb[i] = v;
    }
    for (int i = tid; i < NCODE; i += 64)
        s_cn[i] = cn[i] * SOFT_SCALE;
    __syncthreads();

    const int lane = tid & 31;
    const int wv   = tid >> 5;
    const int half = lane >> 4;
    const int l16  = lane & 15;
    const int kb   = half << 1;

    float* wtw = &s_wt[wv][0];

    // ---- Tile-invariant register caches (explicit, not LICM-dependent) ----
    float cnk[NTILE];
#pragma unroll
    for (int t = 0; t < NTILE; ++t) cnk[t] = s_cn[t * 16 + l16];

    v2f b0c[NTILE], b1c[NTILE];                // pass-1 B fragments (codebook, N-major)
#pragma unroll
    for (int t = 0; t < NTILE; ++t) {
        const float* cr = s_cb + (t * 16 + l16) * CBS;
        b0c[t][0] = cr[kb + 0]; b0c[t][1] = cr[kb + 1];
        b1c[t][0] = cr[kb + 4]; b1c[t][1] = cr[kb + 5];
    }

    v2f bcc[NCHUNK];                           // pass-2 B chunks (codebook, K-major)
    {
        const float* bcr = s_cb + kb * CBS + l16;
#pragma unroll
        for (int c = 0; c < NCHUNK; ++c) {
            bcc[c][0] = bcr[(4 * c) * CBS];
            bcc[c][1] = bcr[(4 * c) * CBS + CBS];
        }
    }
    const float* war = wtw + kb * WSP + l16;   // pass-2 A base (loop-invariant addresses)

    int tile = blockIdx.x * 2 + wv;
    if (tile >= nTiles) return;                // wave-uniform

    // First A fragment (x tile, f32 16x4 layout, K=8 as two chunks)
    v2f a0, a1;
    {
        const float* xr = x + (size_t)(tile * 16 + l16) * 8;
        a0[0] = xr[kb + 0]; a0[1] = xr[kb + 1];
        a1[0] = xr[kb + 4]; a1[1] = xr[kb + 5];
    }

    // NOTE on LDS ordering (no explicit fences needed): DS instructions from a
    // wave are processed in order (ISA 7.3), so pass-2 loads from s_wt observe
    // the pass-1 stores, and the s_st readback observes the staging stores; the
    // compiler cannot reorder them since they may-alias within the same object.
    while (true) {
        const int nextTile = tile + nWaves;
        const bool more = nextTile < nTiles;   // wave-uniform

        // ---- Pass 1: cross via WMMA -> exp2 logits -> W^T to LDS ----
        // logit2 = (2*cross - ||c||^2)*(1/T)*log2e; ||x||^2 cancels; no max needed
        // (max logit2 >= -9.7, |logit2| <~ 90 -> f32 exp2 safe).
#pragma unroll
        for (int t = 0; t < NTILE; ++t) {
            v8f d = {0.f, 0.f, 0.f, 0.f, 0.f, 0.f, 0.f, 0.f};
            d = __builtin_amdgcn_wmma_f32_16x16x4_f32(false, a0, false, b0c[t], (short)0, d, false, false);
            d = __builtin_amdgcn_wmma_f32_16x16x4_f32(false, a1, false, b1c[t], (short)0, d, false, false);
            float e[8];
#pragma unroll
            for (int r = 0; r < 8; ++r)
                e[r] = __builtin_amdgcn_exp2f(fmaf(d[r], 2.0f * SOFT_SCALE, -cnk[t]));
            // lane owns W^T row 16t+l16, cols 8*half..8*half+7: two b128 stores
            float4 lo, hi;
            lo.x = e[0]; lo.y = e[1]; lo.z = e[2]; lo.w = e[3];
            hi.x = e[4]; hi.y = e[5]; hi.z = e[6]; hi.w = e[7];
            float* wcol = wtw + (t * 16 + l16) * WSP + 8 * half;
            *(float4*)(wcol + 0) = lo;
            *(float4*)(wcol + 4) = hi;
        }

        // Prefetch next tile's A fragment (latency hidden under pass 2)
        if (more) {
            const float* xr = x + (size_t)(nextTile * 16 + l16) * 8;
            a0[0] = xr[kb + 0]; a0[1] = xr[kb + 1];
            a1[0] = xr[kb + 4]; a1[1] = xr[kb + 5];
        }

        // ---- Pass 2: quantized = W * C, 60 accumulating WMMAs (K=240) ----
        // D column 8 accumulates the row sums (ones column in B).
        v8f q = {0.f, 0.f, 0.f, 0.f, 0.f, 0.f, 0.f, 0.f};
#pragma unroll
        for (int c = 0; c < NCHUNK; ++c) {
            v2f aw;
            aw[0] = war[(4 * c) * WSP];
            aw[1] = war[(4 * c) * WSP + WSP];
            q = __builtin_amdgcn_wmma_f32_16x16x4_f32(false, aw, false, bcc[c], (short)0, q, false, false);
        }

        // ---- Normalize: 1/rowsum from D column 8 (full-EXEC shuffles), stage, store ----
        float invr[8];
#pragma unroll
        for (int r = 0; r < 8; ++r)
            invr[r] = __builtin_amdgcn_rcpf(__shfl(q[r], (lane & 16) | 8, 32));
        if (l16 < 8) {
            float* st = (float*)&s_st[wv][0];
#pragma unroll
            for (int r = 0; r < 8; ++r)
                st[(r + 8 * half) * 8 + l16] = q[r] * invr[r];
        }
        const float4 o = s_st[wv][lane];       // DS in-order: sees staging stores
        *(float4*)(out + (size_t)tile * 128 + lane * 4) = o;

        if (!more) break;
        tile = nextTile;
    }
}

extern "C" void kernel_launch(void* const* d_in, const int* in_sizes, int n_in,
                              void* d_out, int out_size, void* d_ws, size_t ws_size,
                              hipStream_t stream) {
    const float* x  = (const float*)d_in[0];   // [N, 8]
    const float* cb = (const float*)d_in[1];   // [240, 8]
    const float* cn = (const float*)d_in[2];   // [240]
    float* out = (float*)d_out;                // [N, 8] (forward value == quantized)

    const int n      = in_sizes[0] / 8;        // N = 1,000,000 (multiple of 16)
    const int nTiles = n / 16;                 // 62,500 wave-tiles

    int blocks = (nTiles + 1) / 2;             // 2 persistent waves per block
    if (blocks > 4096) blocks = 4096;          // ~8192 persistent waves, ~7.6 tiles each
    const int nWaves = blocks * 2;

    e8_quant_wmma<<<blocks, 64, 0, stream>>>(x, cb, cn, out, nTiles, nWaves);
}